// FilterDetection_90769838834214
// MI455X (gfx1250) — compile-verified
//
#include <hip/hip_runtime.h>
#include <stdint.h>

#define B_    4
#define N_    32768
#define C_    80
#define K_    2000
#define P_    100
#define NTILE 125            // K_/16
#define PAIRS 7875           // NTILE*(NTILE+1)/2
#define SUPW  64             // u32 words per suppression row (63 used)
#define CAND  2048
#define MAXR  4.1351666f     // |ln(0.016)|

typedef __attribute__((ext_vector_type(2))) float v2f;
typedef __attribute__((ext_vector_type(8))) float v8f;

__device__ __forceinline__ uint32_t flip_f(float f) {
  uint32_t u = __float_as_uint(f);
  return u ^ ((uint32_t)((int32_t)u >> 31) | 0x80000000u);
}
__device__ __forceinline__ float unflip_f(uint32_t o) {
  uint32_t u = (o & 0x80000000u) ? (o ^ 0x80000000u) : ~o;
  return __uint_as_float(u);
}
__device__ __forceinline__ float sat01(float v) {
  return fminf(fmaxf(v, 0.0f), 1.0f);
}

// ---------------- 1. per-anchor score max/argmax + bbox decode + sort key ----
__global__ void prep_kernel(const float* __restrict__ score,
                            const float* __restrict__ logits,
                            const float* __restrict__ regress,
                            const float* __restrict__ anchors,
                            uint64_t* __restrict__ key,
                            float* __restrict__ bbox,
                            uint32_t* __restrict__ label) {
  int idx = blockIdx.x * blockDim.x + threadIdx.x;
  if (idx >= B_ * N_) return;
  int n = idx & (N_ - 1);
  float sc = score[idx];

  const float4* lg = (const float4*)(logits + (size_t)idx * C_);
  float best = -1.0f;
  int bl = 0;
#pragma unroll 4
  for (int c4 = 0; c4 < C_ / 4; ++c4) {
    float4 v = lg[c4];
    float w0 = v.x * sc, w1 = v.y * sc, w2 = v.z * sc, w3 = v.w * sc;
    if (w0 > best) { best = w0; bl = 4 * c4 + 0; }
    if (w1 > best) { best = w1; bl = 4 * c4 + 1; }
    if (w2 > best) { best = w2; bl = 4 * c4 + 2; }
    if (w3 > best) { best = w3; bl = 4 * c4 + 3; }
  }

  float4 a = ((const float4*)anchors)[n];
  float4 r = ((const float4*)regress)[idx];
  float w = a.z - a.x, h = a.w - a.y;
  float cx = a.x + 0.5f * w + r.x * w;
  float cy = a.y + 0.5f * h + r.y * h;
  float dw = fminf(fmaxf(r.z, -MAXR), MAXR);
  float dh = fminf(fmaxf(r.w, -MAXR), MAXR);
  w *= expf(dw);
  h *= expf(dh);
  float4 bb;
  bb.x = sat01(cx - 0.5f * w);
  bb.y = sat01(cy - 0.5f * h);
  bb.z = sat01(cx + 0.5f * w);
  bb.w = sat01(cy + 0.5f * h);
  ((float4*)bbox)[idx] = bb;
  label[idx] = (uint32_t)bl;

  float s = (best > 0.05f) ? best : -1.0f;
  key[idx] = ((uint64_t)flip_f(s) << 15) | (uint64_t)(32767u - (uint32_t)n);
}

// ---------------- 2. init / zero scratch --------------------------------
__global__ void init_kernel(uint32_t* __restrict__ hist,
                            uint64_t* __restrict__ prefix,
                            uint32_t* __restrict__ kk,
                            uint64_t* __restrict__ pivot,
                            uint32_t* __restrict__ ccnt,
                            uint32_t* __restrict__ sup) {
  size_t idx = (size_t)blockIdx.x * blockDim.x + threadIdx.x;
  size_t supWords = (size_t)B_ * K_ * SUPW;
  if (idx < supWords) sup[idx] = 0u;
  if (idx < B_ * 256) hist[idx] = 0u;
  if (idx < B_) {
    prefix[idx] = 0ull;
    pivot[idx] = 0ull;
    kk[idx] = (uint32_t)K_;
    ccnt[idx] = 0u;
  }
}

// ---------------- 3. radix select (6 byte passes over 47-bit key) -------
__global__ void hist_kernel(const uint64_t* __restrict__ key,
                            const uint64_t* __restrict__ prefix,
                            uint32_t* __restrict__ hist, int p) {
  int idx = blockIdx.x * blockDim.x + threadIdx.x;
  if (idx >= B_ * N_) return;
  int b = idx >> 15;
  uint64_t k = key[idx];
  if ((k >> ((p + 1) * 8)) == prefix[b]) {
    uint32_t d = (uint32_t)((k >> (p * 8)) & 0xFFull);
    atomicAdd(&hist[(b << 8) + d], 1u);
  }
}

__global__ void select_kernel(uint64_t* __restrict__ prefix,
                              uint32_t* __restrict__ kk,
                              uint64_t* __restrict__ pivot,
                              uint32_t* __restrict__ hist, int p) {
  int b = blockIdx.x;
  uint32_t* h = &hist[b << 8];
  if (threadIdx.x == 0) {
    uint32_t kr = kk[b];
    int d;
    for (d = 255; d >= 0; --d) {
      uint32_t c = h[d];
      if (kr <= c) break;
      kr -= c;
    }
    if (d < 0) d = 0;
    kk[b] = kr;
    uint64_t pf = (prefix[b] << 8) | (uint64_t)(uint32_t)d;
    prefix[b] = pf;
    if (p == 0) pivot[b] = pf;
  }
  __syncthreads();
  h[threadIdx.x] = 0u;  // ready for next pass
}

__global__ void compact_kernel(const uint64_t* __restrict__ key,
                               const uint64_t* __restrict__ pivot,
                               uint32_t* __restrict__ ccnt,
                               uint64_t* __restrict__ skey) {
  int idx = blockIdx.x * blockDim.x + threadIdx.x;
  if (idx >= B_ * N_) return;
  int b = idx >> 15;
  uint64_t k = key[idx];
  if (k >= pivot[b]) {
    uint32_t pos = atomicAdd(&ccnt[b], 1u);
    if (pos < CAND) skey[(size_t)b * CAND + pos] = k;
  }
}

// ---------------- 4. bitonic sort of 2048 candidate keys + gather -------
__global__ __launch_bounds__(1024) void sort_gather_kernel(
    const uint64_t* __restrict__ skey, const uint32_t* __restrict__ ccnt,
    const float* __restrict__ bbox, const uint32_t* __restrict__ label,
    uint32_t* __restrict__ cidx, float* __restrict__ cscore,
    float* __restrict__ cbox_off, float* __restrict__ cbox_plain) {
  __shared__ uint64_t sh[CAND];
  int b = blockIdx.x, t = threadIdx.x;
  uint32_t cnt = ccnt[b];
  if (cnt > CAND) cnt = CAND;
  for (int i = t; i < CAND; i += 1024)
    sh[i] = (i < (int)cnt) ? skey[(size_t)b * CAND + i] : 0ull;

  for (int k = 2; k <= CAND; k <<= 1) {
    for (int j = k >> 1; j > 0; j >>= 1) {
      __syncthreads();
      for (int i = t; i < CAND; i += 1024) {
        int ixj = i ^ j;
        if (ixj > i) {
          uint64_t x = sh[i], y = sh[ixj];
          bool desc = ((i & k) == 0);
          if (desc ? (x < y) : (x > y)) { sh[i] = y; sh[ixj] = x; }
        }
      }
    }
  }
  __syncthreads();

  for (int i = t; i < K_; i += 1024) {
    uint64_t kv = sh[i];
    uint32_t n = 0;
    float s = -1.0f;
    float4 bp = make_float4(0.f, 0.f, 0.f, 0.f);
    float off = 0.0f;
    if (kv != 0ull) {
      n = 32767u - (uint32_t)(kv & 0x7FFFull);
      s = unflip_f((uint32_t)(kv >> 15));
      bp = ((const float4*)bbox)[(size_t)b * N_ + n];
      off = 2.0f * (float)label[(size_t)b * N_ + n];
    }
    size_t ci = (size_t)b * CAND + i;
    cidx[ci] = n;
    cscore[ci] = s;
    float4 bo = make_float4(bp.x + off, bp.y + off, bp.z + off, bp.w + off);
    ((float4*)cbox_off)[ci] = bo;
    ((float4*)cbox_plain)[ci] = bp;
  }
}

// ---------------- 5. suppression matrix: one wave32 per 16x16 tile ------
// Union denominator area_i + area_j is computed as a rank-2 outer-sum via
// V_WMMA_F32_16X16X4_F32 (matrix pipe co-executes with the VALU min/max).
__global__ __launch_bounds__(32) void iou_sup_kernel(
    const float* __restrict__ cbox_off, uint32_t* __restrict__ sup) {
  __shared__ float sbox[32][4];
  __shared__ uint32_t tb[16];

  int b = blockIdx.x / PAIRS;
  int q = blockIdx.x % PAIRS;
  int it = 0;
  while (q >= NTILE - it) { q -= NTILE - it; ++it; }
  int jt = it + q;

  int lane = threadIdx.x;
  bool isRow = lane < 16;
  int bi = isRow ? (it * 16 + lane) : (jt * 16 + (lane - 16));
  float4 box = ((const float4*)cbox_off)[(size_t)b * CAND + bi];
  float area = (box.z - box.x) * (box.w - box.y);
  sbox[lane][0] = box.x; sbox[lane][1] = box.y;
  sbox[lane][2] = box.z; sbox[lane][3] = box.w;
  if (isRow) tb[lane] = 0u;
  __syncthreads();

  // A[16x4]: col0 = area_row, col1 = 1 ; B[4x16]: row0 = 1, row1 = area_col
  v2f A, Bm;
  A.x = isRow ? area : 0.0f;
  A.y = isRow ? 1.0f : 0.0f;
  Bm.x = isRow ? 1.0f : area;
  Bm.y = 0.0f;
  v8f Cz = {0.f, 0.f, 0.f, 0.f, 0.f, 0.f, 0.f, 0.f};
  v8f D = __builtin_amdgcn_wmma_f32_16x16x4_f32(false, A, false, Bm,
                                                (short)0, Cz, false, false);

  int n = isRow ? lane : (lane - 16);
  float cx1 = sbox[16 + n][0], cy1 = sbox[16 + n][1];
  float cx2 = sbox[16 + n][2], cy2 = sbox[16 + n][3];
  int gj = jt * 16 + n;
  int bitpos = ((jt & 1) << 4) + n;

#pragma unroll
  for (int r = 0; r < 8; ++r) {
    int m = isRow ? r : (8 + r);
    float rx1 = sbox[m][0], ry1 = sbox[m][1];
    float rx2 = sbox[m][2], ry2 = sbox[m][3];
    float iw = fmaxf(0.0f, fminf(rx2, cx2) - fmaxf(rx1, cx1));
    float ih = fmaxf(0.0f, fminf(ry2, cy2) - fmaxf(ry1, cy1));
    float inter = iw * ih;
    float uni = fmaxf(D[r] - inter, 1e-12f);
    int gi = it * 16 + m;
    if ((inter / uni > 0.5f) && (gj > gi)) atomicOr(&tb[m], 1u << bitpos);
  }
  __syncthreads();
  if (isRow) {
    uint32_t wbits = tb[lane];
    if (wbits)
      atomicOr(&sup[((size_t)b * K_ + (it * 16 + lane)) * SUPW + (jt >> 1)],
               wbits);
  }
}

// ---------------- 6. greedy sequential NMS scan over bitmask ------------
__global__ void nms_scan_kernel(const uint32_t* __restrict__ sup,
                                uint32_t* __restrict__ keepw) {
  __shared__ uint32_t kw[63];
  int b = blockIdx.x, t = threadIdx.x;
  if (t < 63) kw[t] = (t < 62) ? 0xFFFFFFFFu : 0x0000FFFFu;  // 2000 bits
  __syncthreads();
  for (int i = 0; i < K_; ++i) {
    uint32_t alive = (kw[i >> 5] >> (i & 31)) & 1u;
    __syncthreads();
    if (alive && t < 63)
      kw[t] &= ~sup[((size_t)b * K_ + i) * SUPW + t];
    __syncthreads();
  }
  if (t < 63) keepw[b * 64 + t] = kw[t];
}

// ---------------- 7. emit first 100 survivors (already score-sorted) ----
__global__ void output_kernel(const uint32_t* __restrict__ keepw,
                              const float* __restrict__ cscore,
                              const uint32_t* __restrict__ cidx,
                              const float* __restrict__ cbox_plain,
                              const float* __restrict__ logits,
                              const float* __restrict__ score,
                              float* __restrict__ out) {
  __shared__ int list[P_];
  __shared__ int cnt_s;
  int b = blockIdx.x, t = threadIdx.x;
  if (t == 0) {
    int cnt = 0;
    for (int k = 0; k < K_ && cnt < P_; ++k) {
      bool kept = (keepw[b * 64 + (k >> 5)] >> (k & 31)) & 1u;
      if (kept && cscore[(size_t)b * CAND + k] > 0.0f) list[cnt++] = k;
    }
    cnt_s = cnt;
  }
  __syncthreads();
  int cnt = cnt_s;
  for (int e = t; e < P_ * C_; e += blockDim.x) {
    int p = e / C_, c = e - p * C_;
    float v = 0.0f;
    if (p < cnt) {
      uint32_t n = cidx[(size_t)b * CAND + list[p]];
      size_t base = (size_t)b * N_ + n;
      v = logits[base * C_ + c] * score[base];
    }
    out[(size_t)b * (P_ * C_) + e] = v;
  }
  for (int e = t; e < P_ * 4; e += blockDim.x) {
    int p = e >> 2;
    float v = (p < cnt)
                  ? cbox_plain[((size_t)b * CAND + list[p]) * 4 + (e & 3)]
                  : 0.0f;
    out[(size_t)B_ * P_ * C_ + (size_t)b * P_ * 4 + e] = v;
  }
}

// ---------------- host launch -------------------------------------------
extern "C" void kernel_launch(void* const* d_in, const int* in_sizes, int n_in,
                              void* d_out, int out_size, void* d_ws,
                              size_t ws_size, hipStream_t stream) {
  const float* score = (const float*)d_in[0];    // [B,N,1]
  const float* logits = (const float*)d_in[1];   // [B,N,C]
  const float* regress = (const float*)d_in[2];  // [B,N,4]
  const float* anchors = (const float*)d_in[3];  // [N,4]
  float* out = (float*)d_out;

  char* w = (char*)d_ws;
  uint64_t* key = (uint64_t*)w;       w += (size_t)B_ * N_ * 8;
  float* bbox = (float*)w;            w += (size_t)B_ * N_ * 4 * 4;
  uint32_t* label = (uint32_t*)w;     w += (size_t)B_ * N_ * 4;
  uint32_t* hist = (uint32_t*)w;      w += (size_t)B_ * 256 * 4;
  uint64_t* prefix = (uint64_t*)w;    w += (size_t)B_ * 8;
  uint64_t* pivot = (uint64_t*)w;     w += (size_t)B_ * 8;
  uint32_t* kk = (uint32_t*)w;        w += (size_t)B_ * 4;
  uint32_t* ccnt = (uint32_t*)w;      w += (size_t)B_ * 4;
  uint64_t* skey = (uint64_t*)w;      w += (size_t)B_ * CAND * 8;
  uint32_t* cidx = (uint32_t*)w;      w += (size_t)B_ * CAND * 4;
  float* cscore = (float*)w;          w += (size_t)B_ * CAND * 4;
  float* cbox_off = (float*)w;        w += (size_t)B_ * CAND * 16;
  float* cbox_plain = (float*)w;      w += (size_t)B_ * CAND * 16;
  uint32_t* sup = (uint32_t*)w;       w += (size_t)B_ * K_ * SUPW * 4;
  uint32_t* keepw = (uint32_t*)w;     w += (size_t)B_ * 64 * 4;

  int nThreads = 256;
  int gridBN = (B_ * N_ + nThreads - 1) / nThreads;  // 512

  init_kernel<<<2048, 256, 0, stream>>>(hist, prefix, kk, pivot, ccnt, sup);
  prep_kernel<<<gridBN, nThreads, 0, stream>>>(score, logits, regress, anchors,
                                               key, bbox, label);
  for (int p = 5; p >= 0; --p) {
    hist_kernel<<<gridBN, nThreads, 0, stream>>>(key, prefix, hist, p);
    select_kernel<<<B_, 256, 0, stream>>>(prefix, kk, pivot, hist, p);
  }
  compact_kernel<<<gridBN, nThreads, 0, stream>>>(key, pivot, ccnt, skey);
  sort_gather_kernel<<<B_, 1024, 0, stream>>>(skey, ccnt, bbox, label, cidx,
                                              cscore, cbox_off, cbox_plain);
  iou_sup_kernel<<<B_ * PAIRS, 32, 0, stream>>>(cbox_off, sup);
  nms_scan_kernel<<<B_, 64, 0, stream>>>(sup, keepw);
  output_kernel<<<B_, 256, 0, stream>>>(keepw, cscore, cidx, cbox_plain,
                                        logits, score, out);
}